// STAR_43765716746651
// MI455X (gfx1250) — compile-verified
//
#include <hip/hip_runtime.h>

#define NVERTS 6890
#define NROWS  20670      // NVERTS*3
#define RPAD   20736      // 54 blocks * 384 rows, padded
#define NB     512
#define NJ     24
#define KD     104        // 10 betas + 93 pose_feat + 1 (template/bias column)
#define KSTEPS 26         // KD/4
#define BT     16         // batches per block (N tile)
#define VT     128        // vertices per block (8 waves x 16 verts)

typedef __attribute__((ext_vector_type(2))) float v2f;
typedef __attribute__((ext_vector_type(8))) float v8f;

__constant__ int c_parent[23] = {0,0,0,1,2,3,4,5,6,7,8,9,9,9,12,13,14,16,17,18,19,20,21};

// ---------------------------------------------------------------------------
// Kernel 0: build transposed/padded coefficient matrix Ct (KD x RPAD):
//   k in [0,10)   -> shapedirs[row][k]
//   k in [10,103) -> posedirs[row][k-10]
//   k == 103      -> v_template[row]   (bias column, paired with Pt[103][b]=1)
// rows >= NROWS are zero. One-time 8.6 MB; makes GEMM loads branch-free+coalesced.
// ---------------------------------------------------------------------------
__global__ void star_coeff(const float* __restrict__ sd, const float* __restrict__ pd,
                           const float* __restrict__ vt, float* __restrict__ Ct) {
    int idx = blockIdx.x * 256 + threadIdx.x;
    if (idx >= KD * RPAD) return;
    int k = idx / RPAD, row = idx % RPAD;
    float v = 0.f;
    if (row < NROWS) {
        if (k < 10)       v = sd[row*10 + k];
        else if (k < 103) v = pd[row*93 + (k - 10)];
        else              v = vt[row];
    }
    Ct[idx] = v;
}

// ---------------------------------------------------------------------------
// Kernel 1: J_aug[j][k][c] = sum_v Jr[j,v] * (c==0 ? v_template[v,k] : shapedirs[v,k,c-1])
// ---------------------------------------------------------------------------
__global__ void star_jaug(const float* __restrict__ Jr, const float* __restrict__ vt,
                          const float* __restrict__ sd, float* __restrict__ JA) {
    __shared__ float red[256];
    int id = blockIdx.x;
    int j = id / 33, rem = id % 33, k = rem / 11, c = rem % 11;
    float s = 0.f;
    for (int v = threadIdx.x; v < NVERTS; v += 256) {
        float coefv = Jr[j*NVERTS + v];
        float x = (c == 0) ? vt[v*3 + k] : sd[(v*3 + k)*10 + (c - 1)];
        s += coefv * x;
    }
    red[threadIdx.x] = s;
    __syncthreads();
    for (int off = 128; off > 0; off >>= 1) {
        if (threadIdx.x < off) red[threadIdx.x] += red[threadIdx.x + off];
        __syncthreads();
    }
    if (threadIdx.x == 0) JA[(j*3 + k)*11 + c] = red[0];
}

// ---------------------------------------------------------------------------
// Kernel 2: per batch (one wave): pose features -> Pt (transposed, KD x NB),
// Rodrigues -> R, joints from J_aug, kinematic chain, G' (3x4 rows) -> Gp.
// ---------------------------------------------------------------------------
__global__ void star_chain(const float* __restrict__ pose, const float* __restrict__ betas,
                           const float* __restrict__ JA, float* __restrict__ Pt,
                           float* __restrict__ Gp) {
    __shared__ float Jl[NJ*3];
    __shared__ float Gl[NJ*12];
    __shared__ float Gg[NJ*12];
    int b = blockIdx.x, t = threadIdx.x;

    if (t < 10) Pt[t*NB + b] = betas[b*10 + t];
    if (t == 0) { Pt[102*NB + b] = betas[b*10 + 1]; Pt[103*NB + b] = 1.f; }

    if (t < NJ) {
        // joint rest positions: affine in betas via J_aug
        #pragma unroll
        for (int k = 0; k < 3; ++k) {
            const float* a = &JA[(t*3 + k)*11];
            float s = a[0];
            #pragma unroll
            for (int n = 0; n < 10; ++n) s += a[1 + n] * betas[b*10 + n];
            Jl[t*3 + k] = s;
        }
        // Rodrigues via quaternion (matches reference eps handling)
        float t0 = pose[b*72 + t*3 + 0];
        float t1 = pose[b*72 + t*3 + 1];
        float t2 = pose[b*72 + t*3 + 2];
        float a0 = t0 + 1e-8f, a1 = t1 + 1e-8f, a2 = t2 + 1e-8f;
        float angle = sqrtf(a0*a0 + a1*a1 + a2*a2);
        float inv = 1.f / angle;
        float n0 = t0*inv, n1 = t1*inv, n2 = t2*inv;
        float hlf = 0.5f*angle;
        float sh = sinf(hlf), ch = cosf(hlf);
        float qw = ch, qx = sh*n0, qy = sh*n1, qz = sh*n2;
        if (t >= 1) {   // pose_feat = pose_quat[:,4:]
            int base = 10 + (t-1)*4;
            Pt[(base+0)*NB + b] = qx;
            Pt[(base+1)*NB + b] = qy;
            Pt[(base+2)*NB + b] = qz;
            Pt[(base+3)*NB + b] = qw - 1.f;
        }
        float rn = rsqrtf(qw*qw + qx*qx + qy*qy + qz*qz);
        qw *= rn; qx *= rn; qy *= rn; qz *= rn;
        float w2=qw*qw, x2=qx*qx, y2=qy*qy, z2=qz*qz;
        float wx=qw*qx, wy=qw*qy, wz=qw*qz;
        float xy=qx*qy, xz=qx*qz, yz=qy*qz;
        float* g = &Gl[t*12];
        g[0] = w2+x2-y2-z2;   g[1] = 2.f*(xy - wz); g[2]  = 2.f*(wy + xz);
        g[4] = 2.f*(wz + xy); g[5] = w2-x2+y2-z2;   g[6]  = 2.f*(yz - wx);
        g[8] = 2.f*(xz - wy); g[9] = 2.f*(wx + yz); g[10] = w2-x2-y2+z2;
    }
    __syncthreads();
    if (t < NJ) {   // relative translation column
        int par = (t == 0) ? 0 : c_parent[t-1];
        #pragma unroll
        for (int k = 0; k < 3; ++k)
            Gl[t*12 + k*4 + 3] = Jl[t*3 + k] - ((t == 0) ? 0.f : Jl[par*3 + k]);
    }
    __syncthreads();
    if (t < 12) Gg[t] = Gl[t];
    __syncthreads();
    // serial kinematic chain; bottom row of every 4x4 is implicitly [0,0,0,1]
    for (int i = 1; i < NJ; ++i) {
        float val = 0.f;
        if (t < 12) {
            int par = c_parent[i-1];
            int r = t >> 2, c = t & 3;
            val = Gg[par*12 + r*4+0]*Gl[i*12 + 0*4 + c]
                + Gg[par*12 + r*4+1]*Gl[i*12 + 1*4 + c]
                + Gg[par*12 + r*4+2]*Gl[i*12 + 2*4 + c];
            if (c == 3) val += Gg[par*12 + r*4 + 3];
        }
        __syncthreads();
        if (t < 12) Gg[i*12 + t] = val;
        __syncthreads();
    }
    if (t < NJ) {   // G' : only col 3 changes (G - G*rest), bottom row stays [0,0,0,1]
        float jx = Jl[t*3+0], jy = Jl[t*3+1], jz = Jl[t*3+2];
        float* go = &Gp[(b*NJ + t)*12];
        #pragma unroll
        for (int r = 0; r < 3; ++r) {
            float g0 = Gg[t*12 + r*4+0], g1 = Gg[t*12 + r*4+1], g2 = Gg[t*12 + r*4+2];
            float g3 = Gg[t*12 + r*4+3] - (g0*jx + g1*jy + g2*jz);
            go[r*4+0] = g0; go[r*4+1] = g1; go[r*4+2] = g2; go[r*4+3] = g3;
        }
    }
}

// ---------------------------------------------------------------------------
// Kernel 3 (fused): WMMA f32 16x16x4 GEMM (v_posed tile, bias folded into K)
// -> LDS -> LBS blend. Block: 16 batches x 128 vertices, 8 waves.
// All GEMM loads are branch-free and coalesced across lanes (k-major layouts).
// ---------------------------------------------------------------------------
__global__ void __launch_bounds__(256)
star_verts(const float* __restrict__ Ct, const float* __restrict__ Pt,
           const float* __restrict__ wgt, const float* __restrict__ trans,
           const float* __restrict__ Gp, float* __restrict__ out) {
    __shared__ float vp[VT*3*BT];     // 384 rows x 16 batches  (24 KB)
    __shared__ float gm[BT*NJ*12];    // 16 batches x 24 joints x 3x4 (18 KB)
    __shared__ float ws_w[VT*NJ];     // 128 verts x 24 weights (12 KB)

    int tid  = threadIdx.x;
    int w    = tid >> 5, lane = tid & 31;
    int half = lane >> 4, ml = lane & 15;
    int bbase = blockIdx.y * BT;
    int vwave = blockIdx.x * VT + w * 16;
    int rowbase = vwave * 3;
    int bcol = bbase + ml;

    // ---- GEMM phase: D(16x16) += A(16x4) * B(4x16), 3 M-tiles per wave ----
    v8f acc[3] = {};
    #pragma unroll
    for (int kk = 0; kk < KSTEPS; ++kk) {
        int ka = kk*4 + 2*half;         // VGPR0 K index for this lane half
        int kb = ka + 1;                // VGPR1 K index
        v2f bm;                         // B[k][n] = Pt[k*NB + n], lanes = n
        bm.x = Pt[ka*NB + bcol];
        bm.y = Pt[kb*NB + bcol];
        #pragma unroll
        for (int m = 0; m < 3; ++m) {   // A[m_idx][k] = Ct[k*RPAD + row], lanes = M
            int row = rowbase + m*16 + ml;
            v2f am;
            am.x = Ct[ka*RPAD + row];
            am.y = Ct[kb*RPAD + row];
            acc[m] = __builtin_amdgcn_wmma_f32_16x16x4_f32(
                false, am, false, bm, (short)0, acc[m], false, false);
        }
    }
    // spill D to LDS: VGPR i holds M = i + 8*half, N = ml (bias already in K=103)
    #pragma unroll
    for (int m = 0; m < 3; ++m) {
        #pragma unroll
        for (int i = 0; i < 8; ++i) {
            int M = i + 8*half;
            vp[(w*48 + m*16 + M)*BT + ml] = acc[m][i];
        }
    }
    // ---- stage transforms and skinning weights ----
    for (int idx = tid; idx < BT*NJ*12; idx += 256)
        gm[idx] = Gp[bbase*NJ*12 + idx];
    for (int idx = tid; idx < VT*NJ; idx += 256) {
        int vl = idx / NJ;
        int v = blockIdx.x * VT + vl;
        ws_w[idx] = (v < NVERTS) ? wgt[v*NJ + (idx % NJ)] : 0.f;
    }
    __syncthreads();

    // ---- blend phase: M(3x4) = sum_j w[v,j] * G'[b,j];  out = M * [p;1] + trans ----
    for (int p = tid; p < VT*BT; p += 256) {
        int vl = p >> 4, n = p & 15;
        int v = blockIdx.x * VT + vl;
        if (v >= NVERTS) continue;
        int b = bbase + n;
        float x = vp[(vl*3+0)*BT + n];
        float y = vp[(vl*3+1)*BT + n];
        float z = vp[(vl*3+2)*BT + n];
        float m00=0,m01=0,m02=0,m03=0,m10=0,m11=0,m12=0,m13=0,m20=0,m21=0,m22=0,m23=0;
        #pragma unroll 4
        for (int j = 0; j < NJ; ++j) {
            float wj = ws_w[vl*NJ + j];
            const float* g = &gm[n*NJ*12 + j*12];
            m00 += wj*g[0];  m01 += wj*g[1];  m02 += wj*g[2];  m03 += wj*g[3];
            m10 += wj*g[4];  m11 += wj*g[5];  m12 += wj*g[6];  m13 += wj*g[7];
            m20 += wj*g[8];  m21 += wj*g[9];  m22 += wj*g[10]; m23 += wj*g[11];
        }
        float ox = m00*x + m01*y + m02*z + m03 + trans[b*3+0];
        float oy = m10*x + m11*y + m12*z + m13 + trans[b*3+1];
        float oz = m20*x + m21*y + m22*z + m23 + trans[b*3+2];
        float* o = &out[(b*NVERTS + v)*3];
        o[0] = ox; o[1] = oy; o[2] = oz;
    }
}

// ---------------------------------------------------------------------------
extern "C" void kernel_launch(void* const* d_in, const int* in_sizes, int n_in,
                              void* d_out, int out_size, void* d_ws, size_t ws_size,
                              hipStream_t stream) {
    (void)in_sizes; (void)n_in; (void)out_size; (void)ws_size;
    const float* pose  = (const float*)d_in[0];
    const float* betas = (const float*)d_in[1];
    const float* trans = (const float*)d_in[2];
    const float* vt    = (const float*)d_in[3];
    const float* sd    = (const float*)d_in[4];
    const float* pd    = (const float*)d_in[5];
    const float* Jr    = (const float*)d_in[6];
    const float* wgt   = (const float*)d_in[7];
    float* out = (float*)d_out;
    float* ws  = (float*)d_ws;
    float* JA = ws;                         // 792 floats
    float* Pt = ws + 1024;                  // KD*NB    = 53248 floats (transposed)
    float* Gp = ws + 1024 + KD*NB;          // NB*NJ*12 = 147456 floats
    float* Ct = ws + 1024 + KD*NB + NB*NJ*12; // KD*RPAD = 2156544 floats (~9 MB total)

    star_coeff<<<dim3((KD*RPAD + 255)/256), dim3(256), 0, stream>>>(sd, pd, vt, Ct);
    star_jaug <<<dim3(NJ*33), dim3(256), 0, stream>>>(Jr, vt, sd, JA);
    star_chain<<<dim3(NB),    dim3(32),  0, stream>>>(pose, betas, JA, Pt, Gp);
    star_verts<<<dim3((NVERTS + VT - 1)/VT, NB/BT), dim3(256), 0, stream>>>(
        Ct, Pt, wgt, trans, Gp, out);
}